// LabelLoss_51866025066625
// MI455X (gfx1250) — compile-verified
//
#include <hip/hip_runtime.h>

typedef float f4  __attribute__((ext_vector_type(4)));
typedef float v2f __attribute__((ext_vector_type(2)));
typedef float v8f __attribute__((ext_vector_type(8)));

#define N_OBJ          16384
#define N_BATCH        256
#define BLK_PER_BATCH  16
#define THREADS        256
#define F4_PER_BATCH   (N_OBJ * 2)                      // 32768 float4 per batch
#define F4_PER_BLOCK   (F4_PER_BATCH / BLK_PER_BATCH)   // 2048
#define F4_PER_THREAD  (F4_PER_BLOCK / THREADS)         // 8

// Stage 1: each block reduces a 32 KB slice of one batch (per tensor) to one partial.
__global__ __launch_bounds__(THREADS) void labelloss_partial(
    const f4* __restrict__ pred, const f4* __restrict__ gt, float* __restrict__ ws) {
  const int slice = blockIdx.x;            // 0..15
  const int b     = blockIdx.y;            // 0..255
  const int t     = threadIdx.x;           // 0..255

  // Flat float4 index. Even index -> channels 0..3 (all counted);
  // odd index -> channels 4..7 (channel 7 == .w masked out).
  // Stride and bases are even, so parity == (t & 1), constant per thread.
  const long base  = (long)b * F4_PER_BATCH + (long)slice * F4_PER_BLOCK + t;
  const float wmask = (t & 1) ? 0.0f : 1.0f;

  float acc = 0.0f;
#pragma unroll
  for (int j = 0; j < F4_PER_THREAD; ++j) {
    // Streamed exactly once: non-temporal so we don't churn L2 with 256 MB.
    f4 p = __builtin_nontemporal_load(&pred[base + (long)j * THREADS]);
    f4 g = __builtin_nontemporal_load(&gt  [base + (long)j * THREADS]);
    float dx = p[0] - g[0];
    float dy = p[1] - g[1];
    float dz = p[2] - g[2];
    float dw = (p[3] - g[3]) * wmask;
    acc = fmaf(dx, dx, acc);
    acc = fmaf(dy, dy, acc);
    acc = fmaf(dz, dz, acc);
    acc = fmaf(dw, dw, acc);
  }

  // Cross-lane wave reduction via the matrix engine, all in fp32:
  // A(16x4) rows: A[m][0]=acc[lane m], A[m][2]=acc[lane m+16], others 0.
  // B(4x16) = all ones  =>  D[m][n] = acc[m] + acc[m+16] for every column n.
  // EXEC is all-1s here (no divergence yet) as WMMA requires.
  v2f a;  a[0] = acc;  a[1] = 0.0f;
  v2f bb; bb[0] = 1.0f; bb[1] = 1.0f;
  v8f c = {};
  v8f d = __builtin_amdgcn_wmma_f32_16x16x4_f32(
      /*neg_a=*/false, a, /*neg_b=*/false, bb,
      /*c_mod=*/(short)0, c, /*reuse_a=*/false, /*reuse_b=*/false);

  // Sum the 8 D VGPRs: lanes 0-15 hold sum over M=0..7, lanes 16-31 over M=8..15.
  float h = d[0] + d[1] + d[2] + d[3] + d[4] + d[5] + d[6] + d[7];
  float tot = h + __shfl_xor(h, 16, 32);   // full 32-lane sum, in every lane

  __shared__ float lds[THREADS / 32];
  const int lane = t & 31;
  const int wave = t >> 5;
  if (lane == 0) lds[wave] = tot;
  __syncthreads();

  if (t == 0) {
    float s = 0.0f;
#pragma unroll
    for (int w = 0; w < THREADS / 32; ++w) s += lds[w];
    ws[b * BLK_PER_BATCH + slice] = s;     // plain store: deterministic across replays
  }
}

// Stage 2: one thread per batch sums its 16 partials and overwrites d_out.
__global__ __launch_bounds__(N_BATCH) void labelloss_finalize(
    const float* __restrict__ ws, float* __restrict__ out) {
  const int b = threadIdx.x;
  float s = 0.0f;
#pragma unroll
  for (int i = 0; i < BLK_PER_BATCH; ++i) s += ws[b * BLK_PER_BATCH + i];
  out[b] = s;
}

extern "C" void kernel_launch(void* const* d_in, const int* in_sizes, int n_in,
                              void* d_out, int out_size, void* d_ws, size_t ws_size,
                              hipStream_t stream) {
  const f4* pred = (const f4*)d_in[0];
  const f4* gt   = (const f4*)d_in[1];
  float*    ws   = (float*)d_ws;     // needs 256*16*4 = 16 KB
  float*    out  = (float*)d_out;

  dim3 grid(BLK_PER_BATCH, N_BATCH);        // 4096 blocks
  labelloss_partial<<<grid, THREADS, 0, stream>>>(pred, gt, ws);
  labelloss_finalize<<<1, N_BATCH, 0, stream>>>(ws, out);
}